// Mamba_19267223290022
// MI455X (gfx1250) — compile-verified
//
#include <hip/hip_runtime.h>
#include <hip/hip_bf16.h>

// ---------------------------------------------------------------------------
// Mamba stack forward for MI455X (gfx1250), f32 end-to-end.
// Dense GEMMs: V_WMMA_F32_16X16X4_F32, A tiles staged into LDS by the Tensor
// Data Mover (tensor_load_to_lds, double-buffered, TDM pad feature used to
// get a 65-DWORD LDS row stride -> bank-conflict-free fragment reads).
// ---------------------------------------------------------------------------

typedef float v2f __attribute__((ext_vector_type(2)));
typedef float v8f __attribute__((ext_vector_type(8)));
typedef unsigned int v4u __attribute__((ext_vector_type(4)));
typedef int v8i_t __attribute__((ext_vector_type(8)));
typedef int v4i_t __attribute__((ext_vector_type(4)));

static constexpr int Bsz = 512;
static constexpr int L   = 41;
static constexpr int D   = 64;
static constexpr int DI  = 128;
static constexpr int Nst = 16;   // state size N
static constexpr int NSk = 3;    // stacks
static constexpr int NLy = 3;    // layers
static constexpr int KC  = 4;    // conv kernel
static constexpr int DTR = 4;
static constexpr int BL  = Bsz * L;   // 20992 rows (multiple of 16)
static constexpr int KB  = 64;        // K-chunk staged in LDS per step
static constexpr int LSTR = KB + 1;   // LDS row stride (TDM pad: +1 DWORD / 64)

// ---------------------------------------------------------------------------
// TDM: one 2D tile load (16 rows x KB cols f32, row stride strideElems) from
// global to LDS. pad_enable inserts 1 DWORD after every 64 DWORDs stored so
// LDS rows land at stride 65 (conflict-free across the 64 LDS banks).
// Descriptor per CDNA5 ISA ch.8; 6-arg builtin on this toolchain.
// ---------------------------------------------------------------------------
__device__ __forceinline__ void tdm_load_tile_f32(unsigned ldsOff,
                                                  const float* gsrc,
                                                  unsigned strideElems) {
  unsigned long long ga = (unsigned long long)(__UINTPTR_TYPE__)gsrc;
  constexpr unsigned rows = 16, cols = KB;
  v4u g0;
  g0[0] = 1u;                                              // count=1, user desc
  g0[1] = ldsOff;                                          // lds_addr (bytes)
  g0[2] = (unsigned)ga;                                    // global_addr[31:0]
  g0[3] = (unsigned)((ga >> 32) & 0x01FFFFFFull)           // global_addr[56:32]
          | (2u << 30);                                    // type = 2 (image)
  v8i_t g1;
  g1[0] = (int)((2u << 16)                 // data_size = 4 bytes
                | (1u << 20)               // pad_enable
                | (5u << 22));             // pad_interval: every 64 DWORDs (+1)
  g1[1] = (int)((cols & 0xFFFFu) << 16);   // tensor_dim0[15:0]
  g1[2] = (int)((cols >> 16) | ((rows & 0xFFFFu) << 16));  // td0 hi | td1 lo
  g1[3] = (int)(((rows >> 16) & 0xFFFFu) | ((cols & 0xFFFFu) << 16)); // | tile_dim0
  g1[4] = (int)(rows & 0xFFFFu);           // tile_dim1 = rows, tile_dim2 = 0
  g1[5] = (int)strideElems;                // tensor_dim0_stride[31:0]
  g1[6] = 0;
  g1[7] = 0;
  v4i_t gz4 = {0, 0, 0, 0};                // groups 2/3 unused (2D tile)
  v8i_t gz8 = {0, 0, 0, 0, 0, 0, 0, 0};
  __builtin_amdgcn_tensor_load_to_lds(g0, g1, gz4, gz4, gz8, 0);
}

// ---------------------------------------------------------------------------
// fp32 WMMA GEMM, shapes compile-time: C[M,NT] = A[M,KT] * W[KT,NT].
// One block (8 waves) per 16-row M-tile; wave w covers N-tiles {w, w+8, ...}.
// A double-buffered in LDS via TDM; B loads fold into immediate offsets.
// EPI: 0 = plain, 1 = +bias, ReLU, 2 = +residual.
// ---------------------------------------------------------------------------
template <int KT, int NT, int EPI>
__global__ void gemm_f32_wmma(const float* __restrict__ A,
                              const float* __restrict__ W,
                              float* __restrict__ C,
                              const float* __restrict__ aux) {
  constexpr int NTILES = NT / 16;
  constexpr int TPW    = (NTILES + 7) / 8;  // n-tiles per wave (<= 3)
  constexpr int NCHUNK = KT / KB;
  __shared__ float smA[2][16 * LSTR];       // 2 x 4.06 KB double buffer
  const int mtile = blockIdx.x;
  const int wave  = threadIdx.x >> 5;
  const int lane  = threadIdx.x & 31;
  const int r  = lane & 15;                 // row-in-tile (A) / col (B,C)
  const int hs = lane >> 4;                 // half select

  const v8f vzero = {0.f, 0.f, 0.f, 0.f, 0.f, 0.f, 0.f, 0.f};
  v8f acc[TPW];
#pragma unroll
  for (int t = 0; t < TPW; ++t) acc[t] = vzero;

  const float* __restrict__ Abase = A + (size_t)mtile * 16 * KT;

  // prologue: stage chunk 0
  if (threadIdx.x < 32) {
    tdm_load_tile_f32((unsigned)(__UINTPTR_TYPE__)&smA[0][0], Abase, KT);
    __builtin_amdgcn_s_wait_tensorcnt(0);
  }
  __syncthreads();

  for (int ch = 0; ch < NCHUNK; ++ch) {
    const int cur = ch & 1;
    if (ch + 1 < NCHUNK && threadIdx.x < 32) {   // prefetch next chunk via TDM
      tdm_load_tile_f32((unsigned)(__UINTPTR_TYPE__)&smA[cur ^ 1][0],
                        Abase + (size_t)(ch + 1) * KB, KT);
    }
    const float* __restrict__ Ap = &smA[cur][0] + r * LSTR + 2 * hs;
#pragma unroll
    for (int t = 0; t < TPW; ++t) {
      const int nt = wave + t * 8;          // wave-uniform -> EXEC all-1s
      if (nt < NTILES) {
        const float* __restrict__ Wp =
            W + (size_t)(ch * KB + 2 * hs) * NT + nt * 16 + r;
        if (ch + 1 < NCHUNK) __builtin_prefetch(Wp + (size_t)KB * NT, 0, 3);
        v8f c = acc[t];
#pragma unroll
        for (int kk = 0; kk < KB; kk += 4) {
          v2f a; a.x = Ap[kk];        a.y = Ap[kk + 1];        // ds, stride 65
          v2f b; b.x = Wp[kk * NT];   b.y = Wp[(kk + 1) * NT]; // imm offsets
          c = __builtin_amdgcn_wmma_f32_16x16x4_f32(
              false, a, false, b, (short)0, c, false, false);
        }
        acc[t] = c;
      }
    }
    __syncthreads();                         // everyone done with smA[cur]
    if (ch + 1 < NCHUNK) {
      if (threadIdx.x < 32) __builtin_amdgcn_s_wait_tensorcnt(0);
      __syncthreads();                       // publish smA[cur^1]
    }
  }

#pragma unroll
  for (int t = 0; t < TPW; ++t) {
    const int nt = wave + t * 8;
    if (nt >= NTILES) continue;
    const int col = nt * 16 + r;
    float* __restrict__ Cp = C + (size_t)(mtile * 16 + 8 * hs) * NT + col;
#pragma unroll
    for (int j = 0; j < 8; ++j) {
      float v = acc[t][j];
      if (EPI == 1) { v += aux[col]; v = v > 0.f ? v : 0.f; }
      if (EPI == 2) { v += aux[(size_t)(mtile * 16 + 8 * hs + j) * NT + col]; }
      Cp[(size_t)j * NT] = v;
    }
  }
}

// ---------------------------------------------------------------------------
// Embedding gather: x[m, d] = embedding[ids[m], d]
// ---------------------------------------------------------------------------
__global__ void embed_kernel(const int* __restrict__ ids,
                             const float* __restrict__ emb,
                             float* __restrict__ x, int total) {
  int idx = blockIdx.x * blockDim.x + threadIdx.x;
  if (idx >= total) return;
  int m = idx >> 6, d = idx & 63;
  x[idx] = emb[ids[m] * D + d];
}

// ---------------------------------------------------------------------------
// RMSNorm over last dim (64): xn = x * rsqrt(mean(x^2)+eps) * w
// ---------------------------------------------------------------------------
__global__ void rmsnorm_kernel(const float* __restrict__ x,
                               const float* __restrict__ w,
                               float* __restrict__ xn, int rows) {
  int m = blockIdx.x * blockDim.x + threadIdx.x;
  if (m >= rows) return;
  const float* xr = x + (size_t)m * D;
  float s = 0.f;
#pragma unroll
  for (int d = 0; d < D; ++d) { float v = xr[d]; s += v * v; }
  float inv = rsqrtf(s * (1.f / D) + 1e-5f);
  float* o = xn + (size_t)m * D;
#pragma unroll
  for (int d = 0; d < D; ++d) o[d] = xr[d] * inv * w[d];
}

// ---------------------------------------------------------------------------
// Depthwise causal conv (K=4) + bias + SiLU.
// ---------------------------------------------------------------------------
__global__ void conv_silu_kernel(const float* __restrict__ ur,
                                 const float* __restrict__ cw,
                                 const float* __restrict__ cb,
                                 float* __restrict__ u2, int total) {
  int idx = blockIdx.x * blockDim.x + threadIdx.x;
  if (idx >= total) return;
  int di = idx & (DI - 1);
  int m  = idx >> 7;          // b*L + l
  int l  = m % L;
  float acc = cb[di];
#pragma unroll
  for (int k = 0; k < KC; ++k) {
    int ls = l - (KC - 1) + k;
    if (ls >= 0)
      acc += ur[(size_t)(m - (KC - 1) + k) * 256 + di] * cw[di * KC + k];
  }
  u2[(size_t)m * DI + di] = acc / (1.f + __expf(-acc));   // silu
}

// ---------------------------------------------------------------------------
// Fold dt projection into x_proj:  Wdt[k, 0:128]  = xp[k,0:4] @ dw
//                                  Wdt[k, 128:160] = xp[k, 4:36]
// ---------------------------------------------------------------------------
__global__ void wdt_prep_kernel(const float* __restrict__ xp,
                                const float* __restrict__ dw,
                                float* __restrict__ Wdt) {
  int idx = blockIdx.x * blockDim.x + threadIdx.x;
  if (idx >= DI * 160) return;
  int k = idx / 160, c = idx % 160;
  float v;
  if (c < DI) {
    v = 0.f;
#pragma unroll
    for (int t = 0; t < DTR; ++t) v += xp[k * 36 + t] * dw[t * DI + c];
  } else {
    v = xp[k * 36 + DTR + (c - DI)];
  }
  Wdt[idx] = v;
}

// ---------------------------------------------------------------------------
// Selective scan. One block per batch element, 128 threads = DI channels.
// ---------------------------------------------------------------------------
__global__ void scan_kernel(const float* __restrict__ dtbc,  // (BL,160)
                            const float* __restrict__ u2,    // (BL,128)
                            const float* __restrict__ ur,    // (BL,256)
                            const float* __restrict__ db,
                            const float* __restrict__ alog,
                            const float* __restrict__ dp,
                            float* __restrict__ yg) {
  int b = blockIdx.x;
  int di = threadIdx.x;
  __shared__ float sm[2 * Nst];     // [Bm(16) | C(16)]
  float a[Nst], h[Nst];
#pragma unroll
  for (int n = 0; n < Nst; ++n) {
    a[n] = -__expf(alog[di * Nst + n]);
    h[n] = 0.f;
  }
  const float dbv = db[di];
  const float dpv = dp[di];
  for (int l = 0; l < L; ++l) {
    int m = b * L + l;
    if (di < 2 * Nst) sm[di] = dtbc[(size_t)m * 160 + DI + di];
    __syncthreads();
    float dtp = dtbc[(size_t)m * 160 + di] + dbv;
    float dt  = dtp > 20.f ? dtp : __logf(1.f + __expf(dtp));   // softplus
    float u   = u2[(size_t)m * DI + di];
    float y   = 0.f;
#pragma unroll
    for (int n = 0; n < Nst; ++n) {
      h[n] = __expf(dt * a[n]) * h[n] + dt * sm[n] * u;
      y += h[n] * sm[Nst + n];
    }
    y += u * dpv;
    float res = ur[(size_t)m * 256 + DI + di];
    yg[(size_t)m * DI + di] = y * (res / (1.f + __expf(-res)));
    __syncthreads();
  }
}

// ---------------------------------------------------------------------------
// Final RMSNorm + softmax-weighted fusion accumulate (per stack s).
// ---------------------------------------------------------------------------
__global__ void fuse_kernel(const float* __restrict__ x,
                            const float* __restrict__ nfw,
                            const float* __restrict__ fw,
                            float* __restrict__ fused, int s, int rows) {
  int m = blockIdx.x * blockDim.x + threadIdx.x;
  if (m >= rows) return;
  float w0 = fw[0], w1 = fw[1], w2 = fw[2];
  float mx = fmaxf(w0, fmaxf(w1, w2));
  float e0 = __expf(w0 - mx), e1 = __expf(w1 - mx), e2 = __expf(w2 - mx);
  float wsf = (s == 0 ? e0 : (s == 1 ? e1 : e2)) / (e0 + e1 + e2);
  const float* xr = x + (size_t)m * D;
  float ssum = 0.f;
#pragma unroll
  for (int d = 0; d < D; ++d) { float v = xr[d]; ssum += v * v; }
  float inv = rsqrtf(ssum * (1.f / D) + 1e-5f);
  float* o = fused + (size_t)m * D;
#pragma unroll
  for (int d = 0; d < D; ++d) {
    float val = xr[d] * inv * nfw[d] * wsf;
    if (s == 0) o[d] = val; else o[d] += val;
  }
}

// ---------------------------------------------------------------------------
// Head: logits = h2 @ W3 + b3 ; out = sigmoid(logits)
// ---------------------------------------------------------------------------
__global__ void head_kernel(const float* __restrict__ h2,
                            const float* __restrict__ W3,
                            const float* __restrict__ b3,
                            float* __restrict__ out) {
  int b = blockIdx.x * blockDim.x + threadIdx.x;
  if (b >= Bsz) return;
  float acc = b3[0];
#pragma unroll
  for (int j = 0; j < 16; ++j) acc += h2[b * 16 + j] * W3[j];
  out[b] = 1.f / (1.f + __expf(-acc));
}

// ---------------------------------------------------------------------------
// Host-side orchestration
// ---------------------------------------------------------------------------
static inline int cdiv(int a, int b) { return (a + b - 1) / b; }

extern "C" void kernel_launch(void* const* d_in, const int* in_sizes, int n_in,
                              void* d_out, int out_size, void* d_ws, size_t ws_size,
                              hipStream_t stream) {
  const int*   ids      = (const int*)  d_in[0];
  const float* embed    = (const float*)d_in[1];
  const float* in_proj  = (const float*)d_in[2];   // (NS,NL,64,256)
  const float* conv_w   = (const float*)d_in[3];   // (NS,NL,128,1,4)
  const float* conv_b   = (const float*)d_in[4];   // (NS,NL,128)
  const float* x_proj   = (const float*)d_in[5];   // (NS,NL,128,36)
  const float* dt_w     = (const float*)d_in[6];   // (NS,NL,4,128)
  const float* dt_b     = (const float*)d_in[7];   // (NS,NL,128)
  const float* A_log    = (const float*)d_in[8];   // (NS,NL,128,16)
  const float* Dp       = (const float*)d_in[9];   // (NS,NL,128)
  const float* out_proj = (const float*)d_in[10];  // (NS,NL,128,64)
  const float* norm_w   = (const float*)d_in[11];  // (NS,NL,64)
  const float* norm_f_w = (const float*)d_in[12];  // (64)
  const float* fusion_w = (const float*)d_in[13];  // (3)
  const float* W1       = (const float*)d_in[14];  // (2624,384)
  const float* b1       = (const float*)d_in[15];
  const float* W2       = (const float*)d_in[16];  // (384,16)
  const float* b2       = (const float*)d_in[17];
  const float* W3       = (const float*)d_in[18];  // (16,1)
  const float* b3       = (const float*)d_in[19];
  float* out = (float*)d_out;

  // ---- scratch layout (floats) ----
  float* ws = (float*)d_ws;
  size_t off = 0;
  float* p_emb  = ws + off; off += (size_t)BL * D;
  float* p_x    = ws + off; off += (size_t)BL * D;
  float* p_xn   = ws + off; off += (size_t)BL * D;
  float* p_ur   = ws + off; off += (size_t)BL * 256;
  float* p_u2   = ws + off; off += (size_t)BL * DI;
  float* p_dtbc = ws + off; off += (size_t)BL * 160;
  float* p_yg   = ws + off; off += (size_t)BL * DI;
  float* p_fus  = ws + off; off += (size_t)BL * D;
  float* p_wdt  = ws + off; off += (size_t)DI * 160;
  float* p_h1   = ws + off; off += (size_t)Bsz * 384;
  float* p_h2   = ws + off; off += (size_t)Bsz * 16;

  const dim3 blk(256);

  // 1) embedding
  embed_kernel<<<cdiv(BL * D, 256), blk, 0, stream>>>(ids, embed, p_emb, BL * D);

  for (int s = 0; s < NSk; ++s) {
    (void)hipMemcpyAsync(p_x, p_emb, (size_t)BL * D * sizeof(float),
                         hipMemcpyDeviceToDevice, stream);
    for (int i = 0; i < NLy; ++i) {
      const int li = s * NLy + i;
      // rmsnorm
      rmsnorm_kernel<<<cdiv(BL, 256), blk, 0, stream>>>(
          p_x, norm_w + (size_t)li * D, p_xn, BL);
      // in_proj GEMM: (BL,64) x (64,256) -> ur
      gemm_f32_wmma<64, 256, 0><<<BL / 16, blk, 0, stream>>>(
          p_xn, in_proj + (size_t)li * D * 2 * DI, p_ur, nullptr);
      // conv + silu
      conv_silu_kernel<<<cdiv(BL * DI, 256), blk, 0, stream>>>(
          p_ur, conv_w + (size_t)li * DI * KC, conv_b + (size_t)li * DI, p_u2,
          BL * DI);
      // fold dt projection into x_proj -> combined weight (128,160)
      wdt_prep_kernel<<<cdiv(DI * 160, 256), blk, 0, stream>>>(
          x_proj + (size_t)li * DI * 36, dt_w + (size_t)li * DTR * DI, p_wdt);
      // dt/B/C GEMM: (BL,128) x (128,160) -> dtbc
      gemm_f32_wmma<128, 160, 0><<<BL / 16, blk, 0, stream>>>(
          p_u2, p_wdt, p_dtbc, nullptr);
      // selective scan + gating
      scan_kernel<<<Bsz, dim3(DI), 0, stream>>>(
          p_dtbc, p_u2, p_ur, dt_b + (size_t)li * DI,
          A_log + (size_t)li * DI * Nst, Dp + (size_t)li * DI, p_yg);
      // out_proj GEMM + residual: x = x + yg @ op
      gemm_f32_wmma<128, 64, 2><<<BL / 16, blk, 0, stream>>>(
          p_yg, out_proj + (size_t)li * DI * D, p_x, p_x);
    }
    // final rmsnorm + softmax-weighted fusion
    fuse_kernel<<<cdiv(BL, 256), blk, 0, stream>>>(
        p_x, norm_f_w, fusion_w, p_fus, s, BL);
  }

  // MLP head
  gemm_f32_wmma<2624, 384, 1><<<Bsz / 16, blk, 0, stream>>>(p_fus, W1, p_h1, b1);
  gemm_f32_wmma<384, 16, 1><<<Bsz / 16, blk, 0, stream>>>(p_h1, W2, p_h2, b2);
  head_kernel<<<cdiv(Bsz, 256), blk, 0, stream>>>(p_h2, W3, b3, out);
}